// SymmetryLoss_35545149342018
// MI455X (gfx1250) — compile-verified
//
#include <hip/hip_runtime.h>

typedef __attribute__((ext_vector_type(2))) float v2f;
typedef __attribute__((ext_vector_type(8))) float v8f;

#define NPLANES   192     // B*P = 64*3
#define NPTILES   12      // 192/16 plane tiles
#define NPOINTS   65536
#define RGRID     128
#define W_R       25.0f

// 1.5 * 2^23 : adding this forces round-to-nearest-even at integer
// granularity; bit pattern is 0x4B400000 + k for k in [-2^22, 2^22).
#define MAGIC_F   12582912.0f
#define MAGIC_I   0x4B400000

__global__ void sym_init_ws(double* __restrict__ acc) {
    acc[0] = 0.0;
}

// Main kernel: sum of distances over all (plane, point) pairs.
// Projection proj[b,p,n] = s.x*nx + s.y*ny + s.z*nz + d is computed with
// V_WMMA_F32_16X16X4_F32: A = 16 points x K4 (x,y,z,1), B = K4 x 16 planes
// (nx,ny,nz,d), D = 16x16 projections (d folded into K=3).
__global__ __launch_bounds__(256)
void sym_sd_kernel(const float* __restrict__ pp,    // (192,4)  n.xyz, d
                   const float* __restrict__ sp,    // (65536,3)
                   const float* __restrict__ grid,  // (128^3,3)
                   const float* __restrict__ gmin,  // (3,)
                   const float* __restrict__ gmax,  // (3,)
                   double* __restrict__ acc)
{
    const int  tid  = threadIdx.x;
    const int  lane = tid & 31;
    const bool hi   = (lane & 16) != 0;   // lanes 16..31
    const int  col  = lane & 15;

    // Normalization: i = clip(round((r - g) * 127/(gmax-g)), 0, 127)
    //              = clamp( as_int(fma(r, s, MAGIC - g*s)) - MAGIC_I , 0, 127 )
    const float g0 = gmin[0], g1 = gmin[1], g2 = gmin[2];
    const float s0 = 127.0f / (gmax[0] - g0);
    const float s1 = 127.0f / (gmax[1] - g1);
    const float s2 = 127.0f / (gmax[2] - g2);
    const float c0 = MAGIC_F - g0 * s0;
    const float c1 = MAGIC_F - g1 * s1;
    const float c2 = MAGIC_F - g2 * s2;

    // Prebuild all 12 B-fragments (planes) + per-lane plane normals for the
    // epilogue. B layout (4x16, 32-bit): lane holds N = lane&15;
    // VGPR0 = K0 (lo lanes) / K2 (hi lanes), VGPR1 = K1 / K3.
    float b0[NPTILES], b1[NPTILES];
    float pnx[NPTILES], pny[NPTILES], pnz[NPTILES];
#pragma unroll
    for (int j = 0; j < NPTILES; ++j) {
        const int q = j * 16 + col;
        const float nx = pp[q*4+0], ny = pp[q*4+1], nz = pp[q*4+2], dd = pp[q*4+3];
        b0[j]  = hi ? nz : nx;
        b1[j]  = hi ? dd : ny;
        pnx[j] = nx; pny[j] = ny; pnz[j] = nz;
    }

    const int wavesPerBlock = blockDim.x >> 5;
    const int waveId = blockIdx.x * wavesPerBlock + (tid >> 5);
    const int nWaves = gridDim.x * wavesPerBlock;
    const int nTiles = NPOINTS / 16;          // 4096 point tiles

    // 8 independent accumulators (one per D-VGPR slot): no serializing FP
    // dependence across the 8 gathered elements of a WMMA tile.
    float accv[8] = {0.f, 0.f, 0.f, 0.f, 0.f, 0.f, 0.f, 0.f};

    for (int t = waveId; t < nTiles; t += nWaves) {
        const int m0 = t * 16;

        // A fragment (16x4, 32-bit): lane holds M = lane&15;
        // VGPR0 = K0 (lo) / K2 (hi), VGPR1 = K1 (lo) / K3 (hi). K3 col = 1.0.
        const int pa = m0 + col;
        const float ax = sp[pa*3+0], ay = sp[pa*3+1], az = sp[pa*3+2];
        v2f a;
        a.x = hi ? az   : ax;
        a.y = hi ? 1.0f : ay;

        // Epilogue point coords: D VGPR v holds point M = v + (hi ? 8 : 0).
        float px[8], py[8], pz[8];
#pragma unroll
        for (int v = 0; v < 8; ++v) {
            const int p = m0 + v + (hi ? 8 : 0);
            px[v] = sp[p*3+0]; py[v] = sp[p*3+1]; pz[v] = sp[p*3+2];
        }

#pragma unroll
        for (int j = 0; j < NPTILES; ++j) {
            v2f b; b.x = b0[j]; b.y = b1[j];
            v8f c = {};
            // D = A x B + 0 : 16 points x 16 planes of proj = s.n + d
            v8f d = __builtin_amdgcn_wmma_f32_16x16x4_f32(
                false, a, false, b, (short)0, c, false, false);

            // Phase 1: reflect + index + issue ALL 8 gathers (no consumption
            // in between -> scheduler can clause the b96 loads for max MLP).
            float rx[8], ry[8], rz[8];
            float cx[8], cy[8], cz[8];
#pragma unroll
            for (int v = 0; v < 8; ++v) {
                const float p2 = d[v] + d[v];
                rx[v] = px[v] - p2 * pnx[j];
                ry[v] = py[v] - p2 * pny[j];
                rz[v] = pz[v] - p2 * pnz[j];
                // round-half-even via magic constant, clamp in integer domain
                int ix = __float_as_int(fmaf(rx[v], s0, c0)) - MAGIC_I;
                int iy = __float_as_int(fmaf(ry[v], s1, c1)) - MAGIC_I;
                int iz = __float_as_int(fmaf(rz[v], s2, c2)) - MAGIC_I;
                ix = min(max(ix, 0), RGRID - 1);
                iy = min(max(iy, 0), RGRID - 1);
                iz = min(max(iz, 0), RGRID - 1);
                const unsigned idx = 3u * (unsigned)((ix * RGRID + iy) * RGRID + iz);
                cx[v] = grid[idx + 0];
                cy[v] = grid[idx + 1];
                cz[v] = grid[idx + 2];
            }
            // Phase 2: consume gathers, distance + raw v_sqrt_f32 (~1 ulp).
#pragma unroll
            for (int v = 0; v < 8; ++v) {
                const float dx = rx[v] - cx[v];
                const float dy = ry[v] - cy[v];
                const float dz = rz[v] - cz[v];
                accv[v] += __builtin_amdgcn_sqrtf(dx*dx + dy*dy + dz*dz);
            }
        }
    }

    const float sum = ((accv[0] + accv[1]) + (accv[2] + accv[3]))
                    + ((accv[4] + accv[5]) + (accv[6] + accv[7]));

    // Block reduction, then one f64 atomic per block for precision.
    __shared__ float red[256];
    red[tid] = sum;
    __syncthreads();
    for (int off = 128; off > 0; off >>= 1) {
        if (tid < off) red[tid] += red[tid + off];
        __syncthreads();
    }
    if (tid == 0) atomicAdd(acc, (double)red[0]);
}

// Regularizer + final combine: one thread per batch element b (64 total).
__global__ __launch_bounds__(64)
void sym_finalize(const float* __restrict__ pp,
                  const double* __restrict__ acc,
                  float* __restrict__ out)
{
    __shared__ float red[64];
    const int b = threadIdx.x;

    float nh[3][3];
#pragma unroll
    for (int p = 0; p < 3; ++p) {
        const float nx = pp[(b*3 + p)*4 + 0];
        const float ny = pp[(b*3 + p)*4 + 1];
        const float nz = pp[(b*3 + p)*4 + 2];
        const float nr = fmaxf(sqrtf(nx*nx + ny*ny + nz*nz), 1e-12f);
        nh[p][0] = nx / nr; nh[p][1] = ny / nr; nh[p][2] = nz / nr;
    }
    float r = 0.0f;
#pragma unroll
    for (int p = 0; p < 3; ++p) {
#pragma unroll
        for (int q = 0; q < 3; ++q) {
            float g = nh[p][0]*nh[q][0] + nh[p][1]*nh[q][1] + nh[p][2]*nh[q][2];
            g -= (p == q) ? 1.0f : 0.0f;
            r += g * g;
        }
    }
    red[b] = r;
    __syncthreads();
    for (int off = 32; off > 0; off >>= 1) {
        if (b < off) red[b] += red[b + off];
        __syncthreads();
    }
    if (b == 0) {
        const float avg_r  = red[0] * (1.0f / 64.0f);
        const float avg_sd = (float)(acc[0] / (64.0 * 65536.0));
        out[0] = avg_sd + W_R * avg_r;
        out[1] = avg_sd;
        out[2] = avg_r;
    }
}

extern "C" void kernel_launch(void* const* d_in, const int* in_sizes, int n_in,
                              void* d_out, int out_size, void* d_ws, size_t ws_size,
                              hipStream_t stream) {
    const float* pp   = (const float*)d_in[0];   // pred_params  (64,3,4)
    const float* sp   = (const float*)d_in[1];   // surface_points (65536,3)
    const float* grid = (const float*)d_in[2];   // closest_point_grid (128^3,3)
    const float* gmin = (const float*)d_in[3];   // (3,)
    const float* gmax = (const float*)d_in[4];   // (3,)
    float*  out = (float*)d_out;
    double* acc = (double*)d_ws;

    sym_init_ws<<<1, 1, 0, stream>>>(acc);
    sym_sd_kernel<<<256, 256, 0, stream>>>(pp, sp, grid, gmin, gmax, acc);
    sym_finalize<<<1, 64, 0, stream>>>(pp, acc, out);
}